// DDNet_75694503625385
// MI455X (gfx1250) — compile-verified
//
#include <hip/hip_runtime.h>
#include <math.h>

// ---------------------------------------------------------------------------
// CDNA5 (gfx1250) fp32 WMMA: V_WMMA_F32_16X16X4_F32
//   A: 16x4  -> 2 VGPR/lane (lane m = lane&15, VGPR j -> K = j + 2*(lane>>4))
//   B: 4x16  -> 2 VGPR/lane (lane n = lane&15, VGPR j -> K = j + 2*(lane>>4))
//   C/D: 16x16 -> 8 VGPR/lane (VGPR v: row = v + 8*(lane>>4), col = lane&15)
// Fragments are staged in LDS "fragment-major": for K-index k, the pair
// (k even, k+1) that one lane consumes is contiguous; the per-fragment slab
// stride is chosen == 32 (mod 64) dwords so the two half-waves hit disjoint
// bank halves -> conflict-free ds_load_b64 per fragment.
// ---------------------------------------------------------------------------
typedef __attribute__((ext_vector_type(2))) float v2f;
typedef __attribute__((ext_vector_type(8))) float v8f;

__device__ __forceinline__ v8f wmma_f32(v2f a, v2f b, v8f c) {
    return __builtin_amdgcn_wmma_f32_16x16x4_f32(false, a, false, b, (short)0, c, false, false);
}
__device__ __forceinline__ v2f lds_frag(const float* p) {
    return *(const v2f*)p;   // 8B-aligned LDS read -> ds_load_b64
}

// ===========================================================================
// Generic GEMM: C[M,N] = A[M,K] @ W[N,K]^T (+bias[N]) (+resid[M,N]) (+GELU)
// Block: 256 threads (8 waves), tile 64(M) x 128(N), K-step 16, LDS double
// buffered. Wave w: rows (w&1)*32, cols (w>>1)*32 (2x2 accumulators).
// A slab: fo = (k>>2)*2 + ((k>>1)&1) in [0,8): As[fo*160 + m*2 + (k&1)]
// B slab:                                      Bs[fo*288 + n*2 + (k&1)]
// (160 % 64 == 32, 288 % 64 == 32 -> conflict-free fragment reads)
// ===========================================================================
#define GT_M 64
#define GT_N 128
#define GT_K 16
#define A_FO 160
#define B_FO 288

__global__ __launch_bounds__(256) void ddnet_gemm_kernel(
    const float* __restrict__ A, const float* __restrict__ W,
    const float* __restrict__ bias, const float* __restrict__ resid,
    float* __restrict__ C, int Mdim, int Ndim, int Kdim, int act)
{
    __shared__ __align__(16) float As[2][8 * A_FO];   // 2 x 5.0 KB
    __shared__ __align__(16) float Bs[2][8 * B_FO];   // 2 x 9.0 KB

    const int tid   = threadIdx.x;
    const int lane  = tid & 31;
    const int wave  = tid >> 5;
    const int lhalf = lane >> 4;
    const int lmod  = lane & 15;
    const int m0    = blockIdx.y * GT_M;
    const int n0    = blockIdx.x * GT_N;
    const int wm    = (wave & 1) * 32;
    const int wn    = (wave >> 1) * 32;
    const int nk    = Kdim >> 4;

    v8f acc[2][2] = {};
    float ra[4], rb[8];

    // ---- prologue: fetch tile 0 to registers, spill to LDS buf 0 ----
    #pragma unroll
    for (int i = 0; i < 4; ++i) {
        int idx = tid + i * 256, kk = idx & 15, mm = idx >> 4;
        ra[i] = A[(size_t)(m0 + mm) * Kdim + kk];
    }
    #pragma unroll
    for (int i = 0; i < 8; ++i) {
        int idx = tid + i * 256, kk = idx & 15, nn = idx >> 4;
        rb[i] = W[(size_t)(n0 + nn) * Kdim + kk];
    }
    #pragma unroll
    for (int i = 0; i < 4; ++i) {
        int idx = tid + i * 256, kk = idx & 15, mm = idx >> 4;
        As[0][((kk >> 2) * 2 + ((kk >> 1) & 1)) * A_FO + mm * 2 + (kk & 1)] = ra[i];
    }
    #pragma unroll
    for (int i = 0; i < 8; ++i) {
        int idx = tid + i * 256, kk = idx & 15, nn = idx >> 4;
        Bs[0][((kk >> 2) * 2 + ((kk >> 1) & 1)) * B_FO + nn * 2 + (kk & 1)] = rb[i];
    }
    __syncthreads();

    for (int kt = 0; kt < nk; ++kt) {
        const int buf = kt & 1;
        // issue next tile's global loads early (hide latency behind WMMA)
        if (kt + 1 < nk) {
            const int k0 = (kt + 1) << 4;
            #pragma unroll
            for (int i = 0; i < 4; ++i) {
                int idx = tid + i * 256, kk = idx & 15, mm = idx >> 4;
                ra[i] = A[(size_t)(m0 + mm) * Kdim + k0 + kk];
            }
            #pragma unroll
            for (int i = 0; i < 8; ++i) {
                int idx = tid + i * 256, kk = idx & 15, nn = idx >> 4;
                rb[i] = W[(size_t)(n0 + nn) * Kdim + k0 + kk];
            }
        }
        const float* Ab = As[buf];
        const float* Bb = Bs[buf];
        #pragma unroll
        for (int t = 0; t < 4; ++t) {
            const int fo = t * 2 + lhalf;
            v2f a0 = lds_frag(&Ab[fo * A_FO + (wm + lmod) * 2]);
            v2f a1 = lds_frag(&Ab[fo * A_FO + (wm + 16 + lmod) * 2]);
            v2f b0 = lds_frag(&Bb[fo * B_FO + (wn + lmod) * 2]);
            v2f b1 = lds_frag(&Bb[fo * B_FO + (wn + 16 + lmod) * 2]);
            acc[0][0] = wmma_f32(a0, b0, acc[0][0]);
            acc[0][1] = wmma_f32(a0, b1, acc[0][1]);
            acc[1][0] = wmma_f32(a1, b0, acc[1][0]);
            acc[1][1] = wmma_f32(a1, b1, acc[1][1]);
        }
        if (kt + 1 < nk) {
            __syncthreads();
            #pragma unroll
            for (int i = 0; i < 4; ++i) {
                int idx = tid + i * 256, kk = idx & 15, mm = idx >> 4;
                As[buf ^ 1][((kk >> 2) * 2 + ((kk >> 1) & 1)) * A_FO + mm * 2 + (kk & 1)] = ra[i];
            }
            #pragma unroll
            for (int i = 0; i < 8; ++i) {
                int idx = tid + i * 256, kk = idx & 15, nn = idx >> 4;
                Bs[buf ^ 1][((kk >> 2) * 2 + ((kk >> 1) & 1)) * B_FO + nn * 2 + (kk & 1)] = rb[i];
            }
            __syncthreads();
        }
    }

    // epilogue: bias / residual / exact GELU
    #pragma unroll
    for (int mi = 0; mi < 2; ++mi)
        #pragma unroll
        for (int ni = 0; ni < 2; ++ni)
            #pragma unroll
            for (int v = 0; v < 8; ++v) {
                int row = m0 + wm + mi * 16 + v + 8 * lhalf;
                int col = n0 + wn + ni * 16 + lmod;
                float val = acc[mi][ni][v];
                if (bias)  val += bias[col];
                if (resid) val += resid[(size_t)row * Ndim + col];
                if (act == 1)
                    val = 0.5f * val * (1.0f + erff(val * 0.70710678118654752f));
                C[(size_t)row * Ndim + col] = val;
            }
}

// ===========================================================================
// Sinusoidal positional encoding: h[b,s,d] += PE(s,d)
// ===========================================================================
__global__ __launch_bounds__(256) void ddnet_pe_kernel(
    float* __restrict__ h, int S, int D)
{
    size_t idx = (size_t)blockIdx.x * 256 + threadIdx.x;
    int d = (int)(idx % D);
    int s = (int)((idx / D) % S);
    int pair = d >> 1;
    float div = __expf(-(float)(2 * pair) * (logf(10000.0f) / (float)D));
    float ang = (float)s * div;
    h[idx] += (d & 1) ? __cosf(ang) : __sinf(ang);
}

// ===========================================================================
// LayerNorm over last dim (D=256): one wave per row, 8 rows per block.
// ===========================================================================
__global__ __launch_bounds__(256) void ddnet_ln_kernel(
    const float* __restrict__ x, const float* __restrict__ gamma,
    const float* __restrict__ beta, float* __restrict__ out, int D)
{
    int row  = blockIdx.x * 8 + (threadIdx.x >> 5);
    int lane = threadIdx.x & 31;
    const float* xr = x + (size_t)row * D;
    float vals[8];
    float s = 0.f;
    #pragma unroll
    for (int i = 0; i < 8; ++i) { vals[i] = xr[lane + i * 32]; s += vals[i]; }
    #pragma unroll
    for (int m = 16; m >= 1; m >>= 1) s += __shfl_xor(s, m, 32);
    float mean = s * (1.0f / 256.0f);
    float vs = 0.f;
    #pragma unroll
    for (int i = 0; i < 8; ++i) { float d = vals[i] - mean; vs += d * d; }
    #pragma unroll
    for (int m = 16; m >= 1; m >>= 1) vs += __shfl_xor(vs, m, 32);
    float rstd = rsqrtf(vs * (1.0f / 256.0f) + 1e-5f);
    float* orow = out + (size_t)row * D;
    #pragma unroll
    for (int i = 0; i < 8; ++i) {
        int c = lane + i * 32;
        orow[c] = (vals[i] - mean) * rstd * gamma[c] + beta[c];
    }
}

// ===========================================================================
// Descriptor layer: one block per position s.
//   Ct[t,g] = cos(2*pi*s / ((2+0.5*t)*1.5^g)), t = i+j in [0,510]
//   hd[b,s,i] = h[b,s,i] + sum_j xt[b,s,j] * sum_g P[i,j,g]*Ct[i+j,g]
// Xt staged fragment-major (slab stride 32 dwords == 32 mod 64: conflict-
// free). Each lane materializes exactly its 8 B-fragment W values in regs.
// ===========================================================================
#define DD 256
#define DG 6

__global__ __launch_bounds__(256) void ddnet_desc_kernel(
    const float* __restrict__ xt, const float* __restrict__ h,
    const float* __restrict__ P, float* __restrict__ hd, int S)
{
    __shared__ __align__(16) float Xs[128 * 32];    // frag-major: 16 KB
    __shared__ float Ct[511 * DG + 1];

    const int s   = blockIdx.x;
    const int tid = threadIdx.x;
    const float two_pi = 6.28318530717958647692f;

    for (int idx = tid; idx < 511 * DG; idx += 256) {
        int t = idx / DG, g = idx % DG;
        float period = (2.0f + 0.5f * (float)t) * powf(1.5f, (float)g);
        Ct[idx] = __cosf(two_pi * (float)s / period);
    }
    for (int idx = tid; idx < 16 * DD; idx += 256) {
        int b = idx >> 8, j = idx & 255;
        Xs[((j >> 2) * 2 + ((j >> 1) & 1)) * 32 + b * 2 + (j & 1)] =
            xt[((size_t)b * S + s) * DD + j];
    }
    __syncthreads();

    const int lane = tid & 31, wave = tid >> 5;
    const int lhalf = lane >> 4, lmod = lane & 15;

    for (int it = wave; it < 16; it += 8) {     // output i-tiles, 2 per wave
        const int i = it * 16 + lmod;           // this lane's B-frag column
        v8f acc = {};
        for (int j0 = 0; j0 < DD; j0 += 16) {
            float wv[4][2];
            #pragma unroll
            for (int t = 0; t < 4; ++t)
                #pragma unroll
                for (int jj = 0; jj < 2; ++jj) {
                    int j = j0 + t * 4 + jj + 2 * lhalf;
                    const float* Pp = P + ((size_t)i * DD + j) * DG;
                    const float* Cp = Ct + (i + j) * DG;
                    float w = 0.f;
                    #pragma unroll
                    for (int g = 0; g < DG; ++g) w += Pp[g] * Cp[g];
                    wv[t][jj] = w;
                }
            #pragma unroll
            for (int t = 0; t < 4; ++t) {
                v2f a = lds_frag(&Xs[((j0 >> 2) + t) * 64 + lhalf * 32 + lmod * 2]);
                v2f b = { wv[t][0], wv[t][1] };
                acc = wmma_f32(a, b, acc);
            }
        }
        #pragma unroll
        for (int v = 0; v < 8; ++v) {
            int b = v + 8 * lhalf;
            size_t off = ((size_t)b * S + s) * DD + it * 16 + lmod;
            hd[off] = acc[v] + h[off];
        }
    }
}

// ===========================================================================
// Flash attention: NH=8, dh=32, scale=1/sqrt(32). 4 waves/block, each wave
// owns one 16-row query tile with online softmax over all 512 keys.
// K/V/P tiles staged fragment-major per wave (slab strides 32 / 96 / 96
// dwords, all == 32 mod 64 -> conflict-free ds_load_b64 fragments).
// ===========================================================================
__global__ __launch_bounds__(128) void ddnet_attn_kernel(
    const float* __restrict__ qkv, float* __restrict__ o, int S)
{
    const int qb    = S / 64;
    const int bh    = blockIdx.x / qb;
    const int qblk  = blockIdx.x % qb;
    const int b     = bh >> 3, head = bh & 7;
    const int tid   = threadIdx.x, lane = tid & 31, wave = tid >> 5;
    const int lhalf = lane >> 4, lmod = lane & 15;
    const int q0    = qblk * 64 + wave * 16;

    __shared__ __align__(16) float Kf[4][16 * 32];  // fo(k-dim dh) x key x 2
    __shared__ __align__(16) float Vf[4][8 * 96];   // fo(k-dim key) x d x 2
    __shared__ __align__(16) float Pf[4][8 * 96];   // fo(k-dim key) x q x 2

    const size_t rs = 768;
    const float* qp = qkv + (size_t)b * S * rs + head * 32;
    const float* kp = qp + 256;
    const float* vp = qp + 512;
    const float scale = 0.17677669529663688f;   // 1/sqrt(32)

    // preload Q A-fragments for all 8 K-steps (dh=32)
    v2f aq[8];
    #pragma unroll
    for (int t = 0; t < 8; ++t) {
        const float* qr = qp + (size_t)(q0 + lmod) * rs;
        aq[t].x = qr[t * 4 + 2 * lhalf];
        aq[t].y = qr[t * 4 + 1 + 2 * lhalf];
    }

    float mrow[8], lrow[8];
    v8f o0 = {}, o1 = {};
    #pragma unroll
    for (int v = 0; v < 8; ++v) { mrow[v] = -1e30f; lrow[v] = 0.f; }

    for (int k0 = 0; k0 < S; k0 += 16) {
        // stage K (frag-major over dh) and V (frag-major over key index)
        for (int i = lane; i < 16 * 32; i += 32) {
            int r = i >> 5, c = i & 31;     // r = key row, c = dh column
            Kf[wave][((c >> 2) * 2 + ((c >> 1) & 1)) * 32 + r * 2 + (c & 1)] =
                kp[(size_t)(k0 + r) * rs + c];
            Vf[wave][((r >> 2) * 2 + ((r >> 1) & 1)) * 96 + c * 2 + (r & 1)] =
                vp[(size_t)(k0 + r) * rs + c];
        }
        __syncthreads();

        // scores 16x16 = Q(16x32) @ K_tile^T
        v8f sc = {};
        #pragma unroll
        for (int t = 0; t < 8; ++t) {
            v2f bk = lds_frag(&Kf[wave][(t * 2 + lhalf) * 32 + lmod * 2]);
            sc = wmma_f32(aq[t], bk, sc);
        }

        // online softmax (rows live in 16-lane halves; xor<=8 stays in half)
        float pr[8];
        #pragma unroll
        for (int v = 0; v < 8; ++v) {
            float sv = sc[v] * scale;
            float mx = sv;
            #pragma unroll
            for (int m = 8; m >= 1; m >>= 1) mx = fmaxf(mx, __shfl_xor(mx, m, 32));
            float mnew = fmaxf(mrow[v], mx);
            float corr = __expf(mrow[v] - mnew);
            float p    = __expf(sv - mnew);
            float rsum = p;
            #pragma unroll
            for (int m = 8; m >= 1; m >>= 1) rsum += __shfl_xor(rsum, m, 32);
            lrow[v] = lrow[v] * corr + rsum;
            mrow[v] = mnew;
            pr[v]   = p;
            o0[v] *= corr;
            o1[v] *= corr;
        }

        // relayout P (D-layout -> A-frag layout) through LDS, then P @ V_tile
        #pragma unroll
        for (int v = 0; v < 8; ++v)
            Pf[wave][((lmod >> 2) * 2 + ((lmod >> 1) & 1)) * 96 +
                     (v + 8 * lhalf) * 2 + (lmod & 1)] = pr[v];
        #pragma unroll
        for (int t = 0; t < 4; ++t) {
            int fo = t * 2 + lhalf;
            v2f ap  = lds_frag(&Pf[wave][fo * 96 + lmod * 2]);
            v2f bv0 = lds_frag(&Vf[wave][fo * 96 + lmod * 2]);
            v2f bv1 = lds_frag(&Vf[wave][fo * 96 + (16 + lmod) * 2]);
            o0 = wmma_f32(ap, bv0, o0);
            o1 = wmma_f32(ap, bv1, o1);
        }
        __syncthreads();
    }

    #pragma unroll
    for (int v = 0; v < 8; ++v) {
        int qr = v + 8 * lhalf;
        float inv = 1.0f / lrow[v];
        size_t base = ((size_t)b * S + q0 + qr) * 256 + head * 32;
        o[base + lmod]      = o0[v] * inv;
        o[base + 16 + lmod] = o1[v] * inv;
    }
}

// ===========================================================================
// Host orchestration
// ===========================================================================
static void launch_gemm(const float* A, const float* W, const float* bias,
                        const float* resid, float* C, int M, int N, int K,
                        int act, hipStream_t stream)
{
    dim3 grid(N / GT_N, M / GT_M);
    ddnet_gemm_kernel<<<grid, 256, 0, stream>>>(A, W, bias, resid, C, M, N, K, act);
}

extern "C" void kernel_launch(void* const* d_in, const int* in_sizes, int n_in,
                              void* d_out, int out_size, void* d_ws, size_t ws_size,
                              hipStream_t stream) {
    (void)in_sizes; (void)n_in; (void)out_size; (void)ws_size;
    const int B = 16, S = 512, IN = 64, D = 256, L = 3, G = 6, DFF = 1024;
    const int Mr = B * S;   // 8192 rows

    const float* x      = (const float*)d_in[0];
    const float* in_w   = (const float*)d_in[1];
    const float* in_b   = (const float*)d_in[2];
    const float* Mw     = (const float*)d_in[3];
    const float* P      = (const float*)d_in[4];
    const float* desc_g = (const float*)d_in[5];
    const float* desc_b = (const float*)d_in[6];
    const float* qkv_w  = (const float*)d_in[7];
    const float* qkv_b  = (const float*)d_in[8];
    const float* out_w  = (const float*)d_in[9];
    const float* out_b  = (const float*)d_in[10];
    const float* ff1_w  = (const float*)d_in[11];
    const float* ff1_b  = (const float*)d_in[12];
    const float* ff2_w  = (const float*)d_in[13];
    const float* ff2_b  = (const float*)d_in[14];
    const float* ln1_g  = (const float*)d_in[15];
    const float* ln1_b  = (const float*)d_in[16];
    const float* ln2_g  = (const float*)d_in[17];
    const float* ln2_b  = (const float*)d_in[18];

    // workspace carve-up (all fp32); total = Mr*(6*D + 3*D + DFF) floats ~109MB
    float* h    = (float*)d_ws;
    float* tmp  = h    + (size_t)Mr * D;
    float* xt   = tmp  + (size_t)Mr * D;
    float* hd   = xt   + (size_t)Mr * D;
    float* qkv  = hd   + (size_t)Mr * D;
    float* obuf = qkv  + (size_t)Mr * 3 * D;
    float* h1   = obuf + (size_t)Mr * D;
    float* ff   = h1   + (size_t)Mr * D;

    // h = x @ in_w^T + in_b ; h += PE
    launch_gemm(x, in_w, in_b, nullptr, h, Mr, D, IN, 0, stream);
    ddnet_pe_kernel<<<(Mr * D) / 256, 256, 0, stream>>>(h, S, D);

    for (int l = 0; l < L; ++l) {
        // ---- descriptor layer ----
        launch_gemm(h, Mw + (size_t)l * D * D, nullptr, nullptr, tmp, Mr, D, D, 0, stream);
        ddnet_ln_kernel<<<Mr / 8, 256, 0, stream>>>(tmp, desc_g + l * D, desc_b + l * D, xt, D);
        ddnet_desc_kernel<<<S, 256, 0, stream>>>(xt, h, P + (size_t)l * D * D * G, hd, S);
        // ---- transformer encoder layer ----
        launch_gemm(hd, qkv_w + (size_t)l * 3 * D * D, qkv_b + (size_t)l * 3 * D,
                    nullptr, qkv, Mr, 3 * D, D, 0, stream);
        ddnet_attn_kernel<<<B * 8 * (S / 64), 128, 0, stream>>>(qkv, obuf, S);
        launch_gemm(obuf, out_w + (size_t)l * D * D, out_b + l * D, hd, tmp, Mr, D, D, 0, stream);
        ddnet_ln_kernel<<<Mr / 8, 256, 0, stream>>>(tmp, ln1_g + l * D, ln1_b + l * D, h1, D);
        launch_gemm(h1, ff1_w + (size_t)l * DFF * D, ff1_b + (size_t)l * DFF,
                    nullptr, ff, Mr, DFF, D, 1 /*gelu*/, stream);
        launch_gemm(ff, ff2_w + (size_t)l * D * DFF, ff2_b + l * D, h1, tmp, Mr, D, DFF, 0, stream);
        float* dst = (l == L - 1) ? (float*)d_out : h;
        ddnet_ln_kernel<<<Mr / 8, 256, 0, stream>>>(tmp, ln2_g + l * D, ln2_b + l * D, dst, D);
    }
}